// GaussRenderer_20770461844021
// MI455X (gfx1250) — compile-verified
//
#include <hip/hip_runtime.h>
#include <math.h>

#define NG     12000
#define CHUNK  64
#define NCHUNK 188            // 188*64 = 12032 >= NG
#define PADN   (NCHUNK*CHUNK) // 12032
#define REC    16             // floats per record (64 bytes)
#define RECB   64
#define IMH    128
#define IMW    128

// record layout (16 floats):
// 0:mx 1:my 2:A(conic00) 3:C(conic11) 4:B(conic01+conic10) 5:opa 6:r 7:g 8:b 9:depth 10:tileMask(bits) 11..15:pad

// ---------------------------------------------------------------------------
// Kernel 1: global stable depth rank + record build, scatter into sorted order
// ---------------------------------------------------------------------------
__global__ __launch_bounds__(256)
void gs_preprocess_sort(const float* __restrict__ means2D,
                        const float* __restrict__ cov2d,
                        const float* __restrict__ color,
                        const float* __restrict__ opacity,
                        const float* __restrict__ depths,
                        float* __restrict__ rec)
{
    __shared__ float sdep[256];
    const int i = blockIdx.x * 256 + threadIdx.x;   // 0..12031 (grid = 47 blocks)
    const float di = (i < NG) ? depths[i] : 3.0e38f;

    int rank = 0;
    for (int base = 0; base < PADN; base += 256) {
        const int j = base + threadIdx.x;
        sdep[threadIdx.x] = (j < NG) ? depths[j] : 3.0e38f;
        __syncthreads();
        if (i < NG) {
            #pragma unroll 4
            for (int k = 0; k < 256; ++k) {
                const float dj = sdep[k];
                const int   jj = base + k;
                if (dj < di || (dj == di && jj < i)) ++rank;
            }
        }
        __syncthreads();
    }

    float4 r0 = {0,0,0,0}, r1 = {0,0,0,0}, r2 = {0,0,0,0}, r3 = {0,0,0,0};
    int slot = i;  // pad records keep their own slot, mask stays 0

    if (i < NG) {
        const float mx  = means2D[2*i+0];
        const float my  = means2D[2*i+1];
        const float c00 = cov2d[4*i+0];
        const float c01 = cov2d[4*i+1];
        const float c10 = cov2d[4*i+2];
        const float c11 = cov2d[4*i+3];

        const float det = c00*c11 - c01*c10;
        const float inv = 1.0f / det;
        const float A   = c11 * inv;                 // conic[0][0]
        const float Cc  = c00 * inv;                 // conic[1][1]
        const float B   = (-c01)*inv + (-c10)*inv;   // conic01+conic10

        const float mid   = 0.5f * (c00 + c11);
        const float s     = sqrtf(fmaxf(mid*mid - det, 0.1f));
        const float radii = 3.0f * ceilf(sqrtf(fmaxf(mid + s, mid - s)));

        const float rminx = fminf(fmaxf(mx - radii, 0.0f), 127.0f);
        const float rminy = fminf(fmaxf(my - radii, 0.0f), 127.0f);
        const float rmaxx = fminf(fmaxf(mx + radii, 0.0f), 127.0f);
        const float rmaxy = fminf(fmaxf(my + radii, 0.0f), 127.0f);

        int m = 0;
        #pragma unroll
        for (int ty = 0; ty < 4; ++ty) {
            #pragma unroll
            for (int tx = 0; tx < 4; ++tx) {
                const float w0  = tx * 32.0f, h0 = ty * 32.0f;
                const float tlx = fmaxf(rminx, w0);
                const float tly = fmaxf(rminy, h0);
                const float brx = fminf(rmaxx, w0 + 31.0f);
                const float bry = fminf(rmaxy, h0 + 31.0f);
                if (brx > tlx && bry > tly) m |= (1 << (ty*4 + tx));
            }
        }

        r0 = make_float4(mx, my, A, Cc);
        r1 = make_float4(B, opacity[i], color[3*i+0], color[3*i+1]);
        r2 = make_float4(color[3*i+2], di, __int_as_float(m), 0.0f);
        slot = rank;
    }

    float4* dst = (float4*)(rec + (size_t)slot * REC);
    dst[0] = r0; dst[1] = r1; dst[2] = r2; dst[3] = r3;
}

// ---------------------------------------------------------------------------
// Kernel 2: per-tile stable compaction of the depth-sorted records
//           (block-wide prefix sum over mask bits; no atomics; deterministic)
// ---------------------------------------------------------------------------
__global__ __launch_bounds__(256)
void gs_compact(const float* __restrict__ rec,
                float* __restrict__ tileRec,
                int* __restrict__ counts)
{
    __shared__ int wsum[8];
    const int t   = blockIdx.x;            // tile 0..15
    const int tid = threadIdx.x;
    const int lane = tid & 31;
    const int wid  = tid >> 5;
    float* __restrict__ dst = tileRec + (size_t)t * PADN * REC;

    int running = 0;
    for (int base = 0; base < PADN; base += 256) {
        const int s = base + tid;
        const int m = __float_as_int(rec[(size_t)s * REC + 10]);
        const int bit = (m >> t) & 1;

        const unsigned mask32 = (unsigned)__ballot(bit);   // wave32
        const int prefix = __popc(mask32 & ((1u << lane) - 1u));
        if (lane == 0) wsum[wid] = __popc(mask32);
        __syncthreads();
        int woff = 0, blocktotal = 0;
        #pragma unroll
        for (int w = 0; w < 8; ++w) {
            const int v = wsum[w];
            if (w < wid) woff += v;
            blocktotal += v;
        }
        if (bit) {
            const float4* srcp = (const float4*)(rec + (size_t)s * REC);
            float4* dstp = (float4*)(dst + (size_t)(running + woff + prefix) * REC);
            dstp[0] = srcp[0]; dstp[1] = srcp[1]; dstp[2] = srcp[2]; dstp[3] = srcp[3];
        }
        running += blocktotal;
        __syncthreads();
    }

    if (tid == 0) counts[t] = running;

    // zero-pad the tail of the last 64-record chunk (alpha=0 no-op records)
    const int padK = ((running + CHUNK - 1) / CHUNK) * CHUNK;
    const float4 z = {0,0,0,0};
    for (int p = running + tid; p < padK; p += 256) {
        float4* dstp = (float4*)(dst + (size_t)p * REC);
        dstp[0] = z; dstp[1] = z; dstp[2] = z; dstp[3] = z;
    }
}

// ---------------------------------------------------------------------------
// Kernel 3: render. One block = 16x16 pixel quadrant of a tile. Streams the
// tile's compacted depth-ordered list through LDS with CDNA5 async-to-LDS
// DMA, double-buffered via s_wait_asynccnt. Inner loop is branch-free.
// ---------------------------------------------------------------------------
__global__ __launch_bounds__(256)
void gs_render(const float* __restrict__ tileRec,
               const int* __restrict__ counts,
               float* __restrict__ out)
{
    __shared__ __align__(16) float buf[2][CHUNK * REC];   // 2 x 4KB

    const int t   = blockIdx.x >> 2;
    const int sub = blockIdx.x & 3;
    const int tx  = t & 3, ty = t >> 2;
    const int lx  = threadIdx.x & 15, ly = threadIdx.x >> 4;
    const int x   = tx * 32 + (sub & 1) * 16 + lx;
    const int y   = ty * 32 + (sub >> 1) * 16 + ly;
    const float px = (float)x, py = (float)y;

    const int K = counts[t];
    const int nchunk = (K + CHUNK - 1) / CHUNK;

    const unsigned lds0 = (unsigned)(unsigned long long)(&buf[0][0]);
    const unsigned lds1 = (unsigned)(unsigned long long)(&buf[1][0]);
    const unsigned lane16 = threadIdx.x * 16u;
    const unsigned long long base =
        (unsigned long long)(tileRec + (size_t)t * PADN * REC);

    float T = 1.0f, acc = 0.0f, cr = 0.0f, cg = 0.0f, cb = 0.0f, dsum = 0.0f;

    if (nchunk > 0) {
        // prologue: async-stage chunk 0 into buf0 (each of 256 lanes moves 16B)
        const unsigned long long src0 = base + lane16;
        asm volatile("global_load_async_to_lds_b128 %0, %1, off"
                     :: "v"(lds0 + lane16), "v"(src0) : "memory");

        for (int c = 0; c < nchunk; ++c) {
            if (c + 1 < nchunk) {
                const unsigned ldsn = ((c + 1) & 1) ? lds1 : lds0;
                const unsigned long long src =
                    base + (unsigned long long)(c + 1) * (CHUNK * RECB) + lane16;
                asm volatile("global_load_async_to_lds_b128 %0, %1, off"
                             :: "v"(ldsn + lane16), "v"(src) : "memory");
                asm volatile("s_wait_asynccnt 1" ::: "memory");
            } else {
                asm volatile("s_wait_asynccnt 0" ::: "memory");
            }
            __syncthreads();

            const float* g = buf[c & 1];
            for (int k = 0; k < CHUNK; ++k) {
                const float* p = g + k * REC;
                const float dx = px - p[0];
                const float dy = py - p[1];
                const float e  = -0.5f * (dx*dx*p[2] + dy*dy*p[3] + dx*dy*p[4]);
                const float gw = expf(e);
                const float alpha = fminf(gw * p[5], 0.99f);
                const float wgt = T * alpha;
                cr   += wgt * p[6];
                cg   += wgt * p[7];
                cb   += wgt * p[8];
                dsum += wgt * p[9];
                acc  += wgt;
                T *= (1.0f - alpha);
            }
            __syncthreads();   // all waves done with buf[c&1] before refill
        }
    }

    const int pid = y * IMW + x;
    const float bg = 1.0f - acc;
    out[pid*3 + 0] = cr + bg;
    out[pid*3 + 1] = cg + bg;
    out[pid*3 + 2] = cb + bg;
    out[IMH*IMW*3 + pid] = dsum;   // depth plane
    out[IMH*IMW*4 + pid] = acc;    // alpha plane
}

extern "C" void kernel_launch(void* const* d_in, const int* in_sizes, int n_in,
                              void* d_out, int out_size, void* d_ws, size_t ws_size,
                              hipStream_t stream) {
    const float* means2D = (const float*)d_in[0];
    const float* cov2d   = (const float*)d_in[1];
    const float* color   = (const float*)d_in[2];
    const float* opacity = (const float*)d_in[3];
    const float* depths  = (const float*)d_in[4];

    char* ws = (char*)d_ws;
    float* rec     = (float*)ws;                                     // 0.77 MB
    float* tileRec = (float*)(ws + (size_t)PADN * RECB);             // 12.3 MB
    int*   counts  = (int*)(ws + (size_t)PADN * RECB * 17);          // 64 B
    float* out = (float*)d_out;

    gs_preprocess_sort<<<PADN / 256, 256, 0, stream>>>(means2D, cov2d, color,
                                                       opacity, depths, rec);
    gs_compact<<<16, 256, 0, stream>>>(rec, tileRec, counts);
    gs_render<<<64, 256, 0, stream>>>(tileRec, counts, out);
}